// GlobalAttentionPooling_48137993454068
// MI455X (gfx1250) — compile-verified
//
#include <hip/hip_runtime.h>
#include <cstddef>

#define C0     32
#define C1     16
#define NGRAPH 1024
#define FDIM   80      // C0 + 3*C1
#define NTS    528     // C0*(C0+1)/2
#define NTSP   544     // NTS padded to a multiple of 32 (uniform loop trip)
#define FTOT   664     // NTS + C1*(C1+1)/2
#define WPB    8       // waves (=nodes) per block in k_logits

typedef __attribute__((ext_vector_type(2))) float v2f;
typedef __attribute__((ext_vector_type(8))) float v8f;

// ---- compile-time packed upper-triangle index table for the 32x32 part ----
// entry p (< NTS): (i<<8)|j for the p-th (i<=j) pair; pad entries -> (0,0)
struct TriuTab { unsigned short ij[NTSP]; };
static constexpr TriuTab make_tab() {
  TriuTab t{}; int p = 0;
  for (int a = 0; a < C0; ++a)
    for (int b = a; b < C0; ++b) { t.ij[p] = (unsigned short)((a << 8) | b); ++p; }
  for (; p < NTSP; ++p) t.ij[p] = 0;
  return t;
}
static constexpr TriuTab kTriu = make_tab();
__constant__ TriuTab g_triu = kTriu;

// branchless SELU: v_exp_f32 + v_cndmask, no exec-mask branching
__device__ __forceinline__ float selu_f(float x) {
  const float scale  = 1.0507009873554805f;
  const float salpha = 1.7580993408473766f;   // scale * alpha
  float e   = __expf(fminf(x, 0.f));
  float neg = salpha * (e - 1.f);
  return x > 0.f ? scale * x : neg;
}

__device__ __forceinline__ int lower_bound_i32(const int* __restrict__ a, int n, int key) {
  int lo = 0, hi = n;
  while (lo < hi) { int mid = (lo + hi) >> 1; if (a[mid] < key) lo = mid + 1; else hi = mid; }
  return lo;
}

// ---------------------------------------------------------------------------
// Kernel 0: zero the softmax denominators
// ---------------------------------------------------------------------------
__global__ void k_zero(float* __restrict__ z, int n) {
  int t = blockIdx.x * blockDim.x + threadIdx.x;
  if (t < n) z[t] = 0.f;
}

// ---------------------------------------------------------------------------
// Kernel 1: wave-per-node attention logits.
//   Gram(v) via V_WMMA_F32_16X16X4_F32, everything else straight-line VALU.
//   ex[node] = exp( selu(tensor_square(node)) . W )
//   z[batch] += ex   (block-aggregated atomics; batch_index is sorted)
// ---------------------------------------------------------------------------
__global__ __launch_bounds__(WPB * 32) void k_logits(
    const float* __restrict__ node_ft,
    const int*   __restrict__ batch_index,
    const float* __restrict__ W,
    float*       __restrict__ ex_out,
    float*       __restrict__ z,
    int n_nodes)
{
  __shared__ float sm[WPB * FDIM];
  __shared__ float sW[FTOT + 1];      // +1: zero pad slot for masked taps
  __shared__ int   sbi[WPB];
  __shared__ float sex[WPB];

  const int lane = threadIdx.x & 31;
  const int wave = threadIdx.x >> 5;
  const int node = blockIdx.x * WPB + wave;
  const bool active = (node < n_nodes);        // wave-uniform

  // stage W into LDS (coalesced, once per block)
  for (int t = threadIdx.x; t < FTOT; t += WPB * 32) sW[t] = W[t];
  if (threadIdx.x == 0) sW[FTOT] = 0.f;

  float* s = sm + wave * FDIM;
  if (active) {
    const float* nf = node_ft + (size_t)node * FDIM;
    s[lane]      = nf[lane];
    s[lane + 32] = nf[lane + 32];
    if (lane < 16) s[lane + 64] = nf[lane + 64];
  }
  __syncthreads();

  if (lane == 0) { sbi[wave] = -1; sex[wave] = 0.f; }

  if (active) {
    const float* v = s + C0;                    // 16 x 3, row-major
    const int cidx = (lane < 16) ? lane : (lane - 16);

    // A: 16x4 f32 (lanes 0..15 -> K=0,1 ; lanes 16..31 -> K=2,3)
    // B: 4x16 f32 (VGPR0: K=0 lanes 0..15 / K=1 lanes 16..31; VGPR1: K=2 / K=3)
    v2f A, B;
    if (lane < 16) {
      A.x = v[cidx * 3 + 0]; A.y = v[cidx * 3 + 1];
      B.x = v[cidx * 3 + 0]; B.y = v[cidx * 3 + 2];
    } else {
      A.x = v[cidx * 3 + 2]; A.y = 0.f;
      B.x = v[cidx * 3 + 1]; B.y = 0.f;
    }

    v8f Cm = {};
    // D = A(16x4) x B(4x16) : per-node Gram matrix v . v^T  (K=3 padded to 4)
    Cm = __builtin_amdgcn_wmma_f32_16x16x4_f32(
        /*neg_a=*/false, A, /*neg_b=*/false, B,
        /*c_mod=*/(short)0, Cm, /*reuse_a=*/false, /*reuse_b=*/false);

    const float rsqrt3 = 0.57735026918962576f;
    float acc = 0.f;

    // C/D layout: VGPR r -> (M=r, N=lane) lanes 0..15; (M=r+8, N=lane-16) else.
    // triu W index = NTS + m*(33-m)/2 + (n-m) = ncol + base(m), base compile-time.
    const int ncol  = cidx;
    const int mbase = (lane < 16) ? 0 : 8;
    const bool lo16 = (lane < 16);
#pragma unroll
    for (int r = 0; r < 8; ++r) {
      const int mA = r, mB = r + 8;
      const int baseA = NTS + (mA * (33 - mA)) / 2 - mA;   // compile-time
      const int baseB = NTS + (mB * (33 - mB)) / 2 - mB;   // compile-time
      int idx = ncol + (lo16 ? baseA : baseB);
      idx = (ncol >= mbase + r) ? idx : FTOT;              // masked -> zero slot
      float t = selu_f(Cm[r] * rsqrt3);
      acc = fmaf(t, sW[idx], acc);
    }

    // 528 scalar-pair features, uniform 17 trips per lane (pads hit zero slot)
#pragma unroll 4
    for (int p = lane; p < NTSP; p += 32) {
      unsigned int ij = g_triu.ij[p];
      float prod = s[ij >> 8] * s[ij & 0xFF];
      int wi = (p < NTS) ? p : FTOT;                       // pad -> zero weight
      acc = fmaf(selu_f(prod), sW[wi], acc);
    }

    // wave32 reduction
#pragma unroll
    for (int off = 16; off > 0; off >>= 1)
      acc += __shfl_xor(acc, off, 32);

    if (lane == 0) {
      float ex = __expf(acc);
      ex_out[node] = ex;
      sbi[wave] = batch_index[node];
      sex[wave] = ex;
    }
  }
  __syncthreads();

  // batch_index is sorted -> merge equal runs, then few atomics per block
  if (threadIdx.x == 0) {
    int i = 0;
    while (i < WPB) {
      int bi = sbi[i];
      if (bi < 0) { ++i; continue; }
      float sum = sex[i];
      int k = i + 1;
      while (k < WPB && sbi[k] == bi) { sum += sex[k]; ++k; }
      atomicAdd(&z[bi], sum);
      i = k;
    }
  }
}

// ---------------------------------------------------------------------------
// Kernel 2: one block per graph; threads 0..79 own channels, segment found by
// binary search on the sorted batch_index; direct (non-atomic) store to out.
// ---------------------------------------------------------------------------
__global__ __launch_bounds__(96) void k_pool(
    const float* __restrict__ node_ft,
    const int*   __restrict__ batch_index,
    const float* __restrict__ ex,
    const float* __restrict__ z,
    float*       __restrict__ out,
    int n_nodes)
{
  __shared__ int srange[2];
  const int g = blockIdx.x;

  if (threadIdx.x == 0) srange[0] = lower_bound_i32(batch_index, n_nodes, g);
  if (threadIdx.x == 1) srange[1] = lower_bound_i32(batch_index, n_nodes, g + 1);
  __syncthreads();

  const int start = srange[0], end = srange[1];
  const float zg   = z[g];
  const float invz = (zg != 0.f) ? (1.f / zg) : 0.f;

  const int c = threadIdx.x;
  if (c < FDIM) {
    float acc = 0.f;
    for (int n = start; n < end; ++n) {
      float w = ex[n] * invz;
      acc = fmaf(node_ft[(size_t)n * FDIM + c], w, acc);
    }
    out[(size_t)g * FDIM + c] = acc;
  }
}

// ---------------------------------------------------------------------------
extern "C" void kernel_launch(void* const* d_in, const int* in_sizes, int n_in,
                              void* d_out, int out_size, void* d_ws, size_t ws_size,
                              hipStream_t stream) {
  const float* node_ft     = (const float*)d_in[0];
  const int*   batch_index = (const int*)d_in[1];
  // d_in[2] = num_graphs scalar (fixed at 1024 by the problem)
  const float* W           = (const float*)d_in[3];
  float*       out         = (float*)d_out;

  const int n_nodes = in_sizes[0] / FDIM;   // 300000

  float* ex = (float*)d_ws;                 // n_nodes floats
  float* z  = ex + n_nodes;                 // NGRAPH floats

  k_zero<<<(NGRAPH + 255) / 256, 256, 0, stream>>>(z, NGRAPH);

  const int nblk = (n_nodes + WPB - 1) / WPB;
  k_logits<<<nblk, WPB * 32, 0, stream>>>(node_ft, batch_index, W, ex, z, n_nodes);

  k_pool<<<NGRAPH, 96, 0, stream>>>(node_ft, batch_index, ex, z, out, n_nodes);
}